// MaskEnhancer_4054449127782
// MI455X (gfx1250) — compile-verified
//
#include <hip/hip_runtime.h>
#include <hip/hip_bf16.h>
#include <math.h>

typedef __attribute__((ext_vector_type(2))) float v2f;
typedef __attribute__((ext_vector_type(8))) float v8f;

#define NBATCH 4
#define NMASK 32
#define C2 1024
#define P2 1024          // 32*32 spatial for layer 2
#define HM 512
#define WM 512

// ---------------------------------------------------------------------------
// Kernel 1: per sampled pixel (32x32 grid), idx = highest mask j (of sam[:,1:])
// containing the pixel, else -1. Also per-mask pixel counts.
// grid = 4 (batch), block = 1024 (one thread per pixel)
// ---------------------------------------------------------------------------
__global__ void k_idx(const int* __restrict__ sam, int* __restrict__ idxb,
                      float* __restrict__ cnt) {
  int b = blockIdx.x;
  int p = threadIdx.x;            // 0..1023
  int y = p >> 5, x = p & 31;
  const int* base = sam + (size_t)b * 33 * HM * WM + (size_t)(y * 16) * WM + x * 16;
  int idx = -1;
  #pragma unroll
  for (int j = 0; j < NMASK; ++j)
    if (base[(size_t)(j + 1) * HM * WM] == 1) idx = j;   // max j wins
  idxb[b * P2 + p] = idx;

  __shared__ int c[NMASK];
  if (threadIdx.x < NMASK) c[threadIdx.x] = 0;
  __syncthreads();
  if (idx >= 0) atomicAdd(&c[idx], 1);
  __syncthreads();
  if (threadIdx.x < NMASK) cnt[b * NMASK + threadIdx.x] = (float)c[threadIdx.x];
}

// ---------------------------------------------------------------------------
// Kernel 2: raw per-mask sums of feat2: raw[b,j,c] = sum_p (idx[b,p]==j) f2[b,c,p]
// grid = B*C = 4096 blocks, 256 threads; LDS float atomics over 32 bins.
// ---------------------------------------------------------------------------
__global__ void k_masksum(const float* __restrict__ feat2, const int* __restrict__ idxb,
                          float* __restrict__ raw) {
  int bc = blockIdx.x;            // 0..4095
  int b = bc >> 10, c = bc & 1023;
  int t = threadIdx.x;            // 256
  __shared__ float bins[NMASK];
  if (t < NMASK) bins[t] = 0.f;
  __syncthreads();
  const float* f = feat2 + ((size_t)b * C2 + c) * P2;
  const int* ib = idxb + b * P2;
  #pragma unroll
  for (int r = 0; r < 4; ++r) {
    int p = t + 256 * r;
    int j = ib[p];
    if (j >= 0) atomicAdd(&bins[j], f[p]);   // ds_add_f32
  }
  __syncthreads();
  if (t < NMASK) raw[((size_t)b * NMASK + t) * C2 + c] = bins[t];
}

// ---------------------------------------------------------------------------
// Kernel 3: mfn[b,j,:] = l2norm(raw[b,j,:] / (cnt+1e-5))
// grid = B*M = 128 blocks, 256 threads
// ---------------------------------------------------------------------------
__global__ void k_norm(const float* __restrict__ raw, const float* __restrict__ cnt,
                       float* __restrict__ mfn) {
  int bj = blockIdx.x;            // 0..127
  int b = bj >> 5, j = bj & 31;
  int t = threadIdx.x;
  const float* r = raw + (size_t)bj * C2;
  float inv = 1.f / (cnt[b * NMASK + j] + 1e-5f);
  __shared__ float red[256];
  float mv[4];
  float ss = 0.f;
  #pragma unroll
  for (int k = 0; k < 4; ++k) {
    float m = r[t + 256 * k] * inv;
    mv[k] = m;
    ss += m * m;
  }
  red[t] = ss; __syncthreads();
  for (int off = 128; off > 0; off >>= 1) {
    if (t < off) red[t] += red[t + off];
    __syncthreads();
  }
  float scale = 1.f / fmaxf(sqrtf(red[0]), 1e-12f);
  float* o = mfn + (size_t)bj * C2;
  #pragma unroll
  for (int k = 0; k < 4; ++k) o[t + 256 * k] = mv[k] * scale;
}

// ---------------------------------------------------------------------------
// Kernel 4: q[b] = mfn[b] (32x1024) @ W^T (1024x1024) + bias, f32 WMMA 16x16x4.
// One wave (32 threads) per (b, mtile in {0,1}, ntile in 0..63): 512 waves.
// A layout (16x4 f32): lanes 0-15 -> M=lane, K={0,1} in v[0],v[1];
//                      lanes 16-31 -> M=lane-16, K={2,3}.
// B layout (4x16 f32): lanes 0-15 -> N=lane, K={0,1}; lanes 16-31 -> K={2,3}.
// B[k,n] = W[n,k], so B loads mirror A loads on rows of W.
// C/D (16x16 f32): VGPR r: lanes 0-15 -> M=r, lanes 16-31 -> M=8+r; N=lane%16.
// ---------------------------------------------------------------------------
__global__ void k_gemm_q(const float* __restrict__ mfn, const float* __restrict__ W,
                         const float* __restrict__ bias, float* __restrict__ q) {
  int id = blockIdx.x;            // 0..511
  int b  = id >> 7;
  int mt = (id >> 6) & 1;
  int nt = id & 63;
  int lane = threadIdx.x;         // 0..31
  int lh = lane >> 4, lm = lane & 15;
  const float* A  = mfn + ((size_t)b * NMASK + mt * 16 + lm) * C2 + 2 * lh;
  const float* Bm = W   + ((size_t)(nt * 16 + lm)) * C2 + 2 * lh;
  v8f acc = {};
  for (int k = 0; k < C2; k += 4) {
    v2f a  = *(const v2f*)(A + k);
    v2f bb = *(const v2f*)(Bm + k);
    acc = __builtin_amdgcn_wmma_f32_16x16x4_f32(
        /*neg_a=*/false, a, /*neg_b=*/false, bb,
        /*c_mod=*/(short)0, acc, /*reuse_a=*/false, /*reuse_b=*/false);
  }
  float bv = bias[nt * 16 + lm];
  float* qo = q + (size_t)b * NMASK * C2 + (size_t)nt * 16 + lm;
  #pragma unroll
  for (int r = 0; r < 8; ++r) {
    int row = mt * 16 + (lh ? 8 + r : r);
    qo[(size_t)row * C2] = acc[r] + bv;
  }
}

// ---------------------------------------------------------------------------
// Kernel 5: graph attention tail, one block per batch, 256 threads.
// sim = qq^T/(n_i n_j + 1e-8); edge = sim/(rowsum+1e-8); gp = edge @ q;
// mf2 = mfn + sigmoid(g) * l2norm(gp)
// ---------------------------------------------------------------------------
__global__ void k_attn(const float* __restrict__ q, const float* __restrict__ mfn,
                       const float* __restrict__ gptr, float* __restrict__ mf2) {
  int b = blockIdx.x;
  int t = threadIdx.x;            // 256
  const float* qb = q + (size_t)b * NMASK * C2;
  __shared__ float red[256];
  __shared__ float nrm[NMASK];
  __shared__ float sim[NMASK * NMASK];
  __shared__ float rs[NMASK];
  __shared__ float gprow[C2];
  // row norms of q
  for (int i = 0; i < NMASK; ++i) {
    float s = 0.f;
    for (int c = t; c < C2; c += 256) { float v = qb[i * C2 + c]; s += v * v; }
    red[t] = s; __syncthreads();
    for (int off = 128; off > 0; off >>= 1) {
      if (t < off) red[t] += red[t + off];
      __syncthreads();
    }
    if (t == 0) nrm[i] = sqrtf(red[0]);
    __syncthreads();
  }
  // cosine similarity, 4 pairs per thread
  #pragma unroll
  for (int k = 0; k < 4; ++k) {
    int pp = t + 256 * k;
    int i = pp >> 5, j = pp & 31;
    float d = 0.f;
    for (int c = 0; c < C2; ++c) d += qb[i * C2 + c] * qb[j * C2 + c];
    sim[pp] = d / (nrm[i] * nrm[j] + 1e-8f);
  }
  __syncthreads();
  if (t < NMASK) {
    float s = 0.f;
    #pragma unroll
    for (int j = 0; j < NMASK; ++j) s += sim[t * NMASK + j];
    rs[t] = s + 1e-8f;
  }
  __syncthreads();
  float sg = 1.f / (1.f + expf(-gptr[0]));
  for (int i = 0; i < NMASK; ++i) {
    float invr = 1.f / rs[i];
    float ssl = 0.f;
    for (int c = t; c < C2; c += 256) {
      float g = 0.f;
      #pragma unroll
      for (int j = 0; j < NMASK; ++j) g += sim[i * NMASK + j] * qb[j * C2 + c];
      g *= invr;
      gprow[c] = g;
      ssl += g * g;
    }
    red[t] = ssl; __syncthreads();
    for (int off = 128; off > 0; off >>= 1) {
      if (t < off) red[t] += red[t + off];
      __syncthreads();
    }
    float scale = 1.f / fmaxf(sqrtf(red[0]), 1e-12f);
    for (int c = t; c < C2; c += 256)
      mf2[((size_t)b * NMASK + i) * C2 + c] =
          mfn[((size_t)b * NMASK + i) * C2 + c] + sg * scale * gprow[c];
    __syncthreads();
  }
}

// ---------------------------------------------------------------------------
// Kernel 6: out2 = feat2 + scatter(mf2). Masks disjoint -> direct gather by idx.
// ---------------------------------------------------------------------------
__global__ void k_out2(const float* __restrict__ feat2, const int* __restrict__ idxb,
                       const float* __restrict__ mf2, float* __restrict__ out2) {
  int gid = blockIdx.x * 256 + threadIdx.x;   // 0 .. 4M-1
  int p = gid & 1023;
  int c = (gid >> 10) & 1023;
  int b = gid >> 20;
  int j = idxb[b * P2 + p];
  float add = (j >= 0) ? mf2[((size_t)b * NMASK + j) * C2 + c] : 0.f;
  out2[gid] = feat2[gid] + add;
}

extern "C" void kernel_launch(void* const* d_in, const int* in_sizes, int n_in,
                              void* d_out, int out_size, void* d_ws, size_t ws_size,
                              hipStream_t stream) {
  // dict order: feat0, feat1, feat2, sam_masks, W0,b0,g0, W1,b1,g1, W2,b2,g2
  const float* feat0 = (const float*)d_in[0];
  const float* feat1 = (const float*)d_in[1];
  const float* feat2 = (const float*)d_in[2];
  const int*   sam   = (const int*)d_in[3];
  const float* W2    = (const float*)d_in[10];
  const float* b2    = (const float*)d_in[11];
  const float* g2    = (const float*)d_in[12];
  float* out = (float*)d_out;

  const size_t N0 = (size_t)4 * 256 * 128 * 128;   // 16,777,216
  const size_t N1 = (size_t)4 * 512 * 64 * 64;     //  8,388,608
  // layers 0 and 1 are identity in the reference: plain D2D copies
  hipMemcpyAsync(out,      feat0, N0 * sizeof(float), hipMemcpyDeviceToDevice, stream);
  hipMemcpyAsync(out + N0, feat1, N1 * sizeof(float), hipMemcpyDeviceToDevice, stream);

  char* ws = (char*)d_ws;
  int*   idxb = (int*)(ws);                                  // 16 KB
  float* cnt  = (float*)(ws + (64 << 10));                   // 0.5 KB
  float* raw  = (float*)(ws + (128 << 10));                  // 512 KB
  float* mfn  = (float*)(ws + (128 << 10) + (512 << 10));    // 512 KB
  float* qbuf = (float*)(ws + (128 << 10) + (1024 << 10));   // 512 KB
  float* mf2  = (float*)(ws + (128 << 10) + (1536 << 10));   // 512 KB

  k_idx    <<<NBATCH, 1024, 0, stream>>>(sam, idxb, cnt);
  k_masksum<<<NBATCH * C2, 256, 0, stream>>>(feat2, idxb, raw);
  k_norm   <<<NBATCH * NMASK, 256, 0, stream>>>(raw, cnt, mfn);
  k_gemm_q <<<NBATCH * 2 * 64, 32, 0, stream>>>(mfn, W2, b2, qbuf);
  k_attn   <<<NBATCH, 256, 0, stream>>>(qbuf, mfn, g2, mf2);
  k_out2   <<<(NBATCH * C2 * P2) / 256, 256, 0, stream>>>(feat2, idxb, mf2, out + N0 + N1);
}